// ArmaNet01_60687887893294
// MI455X (gfx1250) — compile-verified
//
#include <hip/hip_runtime.h>
#include <math.h>
#include <stdint.h>

// ARMAConv (K=3 stacks, T=4 layers, F_IN=F_OUT=1) on a 1M-node / 32M-edge graph.
// All "matmuls" are 1x1 scalars -> SpMV-style gather/scatter workload.
//  - K=3 stack values packed as float4/node: 1 b128 L2 gather per edge.
//  - Edge streams staged through LDS with the CDNA5 Tensor Data Mover
//    (tensor_load_to_lds + s_wait_tensorcnt), double-buffered, so the wave's
//    vector-load path is dedicated to the random gathers/atomics.
//  - NT temporal hints + global_prefetch_b8 on the remaining streaming passes.

static constexpr int THREADS = 256;
static constexpr int EDGE_BLOCKS = 8192;
static constexpr int TILE = 1024;   // edges per TDM tile (LDS: 2*(8K+8K+4K)=40KB)

#define GSTART ((long long)blockIdx.x * blockDim.x + threadIdx.x)
#define GSTRIDE ((long long)gridDim.x * blockDim.x)

__global__ void arma_zero_kernel(float* __restrict__ p, long long n) {
    for (long long i = GSTART; i < n; i += GSTRIDE) p[i] = 0.0f;
}

// deg[col_e] += w_e
__global__ void arma_deg_kernel(const long long* __restrict__ col,
                                const float* __restrict__ w,
                                float* __restrict__ deg, long long E) {
    const long long gs = GSTRIDE;
    for (long long e = GSTART; e < E; e += gs) {
        long long nxt = e + gs;
        if (nxt < E) {
            __builtin_prefetch(col + nxt, 0, 1);
            __builtin_prefetch(w + nxt, 0, 1);
        }
        int c = (int)__builtin_nontemporal_load(col + e);
        float we = __builtin_nontemporal_load(w + e);
        atomicAdd(deg + c, we);
    }
}

// deg -> dinv in place
__global__ void arma_dinv_kernel(float* __restrict__ d, int n) {
    for (long long i = GSTART; i < n; i += GSTRIDE) {
        float v = d[i];
        d[i] = (v > 0.0f) ? __frsqrt_rn(v) : 0.0f;
    }
}

// px[col] += norm_e * x[row]   (layer-0 propagate; shared across K stacks)
__global__ void arma_px_kernel(const long long* __restrict__ row,
                               const long long* __restrict__ col,
                               const float* __restrict__ w,
                               const float* __restrict__ dinv,
                               const float* __restrict__ x,
                               float* __restrict__ px, long long E) {
    const long long gs = GSTRIDE;
    for (long long e = GSTART; e < E; e += gs) {
        long long nxt = e + gs;
        if (nxt < E) {
            __builtin_prefetch(row + nxt, 0, 1);
            __builtin_prefetch(col + nxt, 0, 1);
            __builtin_prefetch(w + nxt, 0, 1);
        }
        int r = (int)__builtin_nontemporal_load(row + e);
        int c = (int)__builtin_nontemporal_load(col + e);
        float we = __builtin_nontemporal_load(w + e);
        float nrm = dinv[r] * we * dinv[c];
        atomicAdd(px + c, nrm * x[r]);
    }
}

// cur[i].k = relu(W0[k]*px[i] + V0[k]*x[i] + b0[k])
__global__ void arma_layer0_kernel(const float* __restrict__ x,
                                   const float* __restrict__ px,
                                   const float* __restrict__ W0,
                                   const float* __restrict__ V0,
                                   const float* __restrict__ b0,
                                   float4* __restrict__ cur, int n) {
    for (long long i = GSTART; i < n; i += GSTRIDE) {
        float xi = x[i], p = px[i];
        float4 o;
        o.x = fmaxf(fmaf(W0[0], p, fmaf(V0[0], xi, b0[0])), 0.0f);
        o.y = fmaxf(fmaf(W0[1], p, fmaf(V0[1], xi, b0[1])), 0.0f);
        o.z = fmaxf(fmaf(W0[2], p, fmaf(V0[2], xi, b0[2])), 0.0f);
        o.w = 0.0f;
        cur[i] = o;
    }
}

#if defined(__HIP_DEVICE_COMPILE__) && defined(__has_builtin)
#if __has_builtin(__builtin_amdgcn_tensor_load_to_lds) && \
    __has_builtin(__builtin_amdgcn_s_wait_tensorcnt)
#define ARMA_USE_TDM 1
#endif
#endif

#if defined(ARMA_USE_TDM)
typedef unsigned int tdm_u32x4 __attribute__((ext_vector_type(4)));
typedef int          tdm_i32x8 __attribute__((ext_vector_type(8)));
typedef int          tdm_i32x4 __attribute__((ext_vector_type(4)));

// 1-D TDM copy: nelem elements of (1<<dsz_code) bytes, global -> LDS.
// D# per CDNA5 ISA ch.8: group0 {count=1, lds_addr, global_addr, type=2},
// group1 {data_size, tensor_dim0 = tile_dim0 = nelem, dim0_stride = nelem},
// groups 2/3 (and the trailing group) zero: <=2-D tensor, no iterate/gather.
// This toolchain exposes the 6-arg builtin: (u32x4, i32x8, i32x4, i32x4, i32x8, cpol).
__device__ __forceinline__ void tdm_load_1d(const void* lds_ptr, const void* gptr,
                                            unsigned int nelem, unsigned int dsz_code) {
    unsigned long long ga = (unsigned long long)(uintptr_t)gptr;
    unsigned int lds_addr = (unsigned int)(uintptr_t)lds_ptr;  // addr[31:0] = LDS byte offset
    tdm_u32x4 g0;
    g0[0] = 1u;                                   // count=1 (valid user descriptor)
    g0[1] = lds_addr;                             // lds_addr
    g0[2] = (unsigned int)ga;                     // global_addr[31:0]
    g0[3] = (unsigned int)(ga >> 32) | (2u << 30);// global_addr[56:32] | type=2
    tdm_i32x8 g1;
    g1[0] = (int)(dsz_code << 16);                // wg_mask=0, data_size
    g1[1] = (int)((nelem & 0xFFFFu) << 16);       // tensor_dim0[15:0] in bits 63:48
    g1[2] = (int)((nelem >> 16) | (1u << 16));    // tensor_dim0[31:16] | tensor_dim1=1
    g1[3] = (int)((nelem & 0xFFFFu) << 16);       // tile_dim0 in bits 127:112
    g1[4] = 0;                                    // tile_dim1=0, tile_dim2=0 (1-D tile)
    g1[5] = (int)nelem;                           // tensor_dim0_stride[31:0]
    g1[6] = 0;
    g1[7] = 0;
    tdm_i32x4 gz4; gz4[0] = 0; gz4[1] = 0; gz4[2] = 0; gz4[3] = 0;
    tdm_i32x8 gz8;
    gz8[0] = 0; gz8[1] = 0; gz8[2] = 0; gz8[3] = 0;
    gz8[4] = 0; gz8[5] = 0; gz8[6] = 0; gz8[7] = 0;
    __builtin_amdgcn_tensor_load_to_lds(g0, g1, gz4, gz4, gz8, 0);
}
#endif

// acc[col].k += norm_e * cur[row].k, k=0..2.  Edge streams staged via TDM->LDS,
// double-buffered; random gathers/atomics stay on the vector path (L2-resident).
__global__ __launch_bounds__(THREADS) void arma_spmv3_kernel(
        const long long* __restrict__ row, const long long* __restrict__ col,
        const float* __restrict__ w, const float* __restrict__ dinv,
        const float4* __restrict__ cur, float* __restrict__ acc, long long E) {
    __shared__ long long sRow[2][TILE];
    __shared__ long long sCol[2][TILE];
    __shared__ float     sW[2][TILE];

    long long span = (E + gridDim.x - 1) / gridDim.x;
    long long beg = (long long)blockIdx.x * span;
    long long end = beg + span; if (end > E) end = E;
    if (beg >= end) return;
    long long nt = (end - beg + TILE - 1) / TILE;
    const bool leader = (threadIdx.x < 32);   // wave 0 drives the TDM pipeline

#if defined(ARMA_USE_TDM)
    if (leader) {
        unsigned int c0 = (unsigned int)((end - beg < TILE) ? (end - beg) : TILE);
        tdm_load_1d(&sRow[0][0], row + beg, c0, 3);   // 8B elems
        tdm_load_1d(&sCol[0][0], col + beg, c0, 3);
        tdm_load_1d(&sW[0][0],   w   + beg, c0, 2);   // 4B elems
    }
#endif
    for (long long t = 0; t < nt; ++t) {
        int par = (int)(t & 1);
        long long tb = beg + t * TILE;
        int cnt = (int)(((end - tb) < TILE) ? (end - tb) : TILE);
#if defined(ARMA_USE_TDM)
        if (leader) {
            if (t + 1 < nt) {
                long long nb = tb + TILE;
                unsigned int c1 = (unsigned int)(((end - nb) < TILE) ? (end - nb) : TILE);
                tdm_load_1d(&sRow[1 - par][0], row + nb, c1, 3);
                tdm_load_1d(&sCol[1 - par][0], col + nb, c1, 3);
                tdm_load_1d(&sW[1 - par][0],   w   + nb, c1, 2);
                __builtin_amdgcn_s_wait_tensorcnt(3);  // tile t's 3 ops done (in-order)
            } else {
                __builtin_amdgcn_s_wait_tensorcnt(0);
            }
        }
#else
        // Fallback: cooperative vector loads into the same LDS tile.
        for (int i = threadIdx.x; i < cnt; i += blockDim.x) {
            sRow[par][i] = row[tb + i];
            sCol[par][i] = col[tb + i];
            sW[par][i]   = w[tb + i];
        }
#endif
        __syncthreads();   // LDS tile t visible to all waves
        for (int i = threadIdx.x; i < cnt; i += blockDim.x) {
            int r = (int)sRow[par][i];
            int c = (int)sCol[par][i];
            float nrm = dinv[r] * sW[par][i] * dinv[c];
            float4 g = cur[r];                  // L2-resident b128 gather
            float* a = acc + 4ll * c;
            atomicAdd(a + 0, nrm * g.x);
            atomicAdd(a + 1, nrm * g.y);
            atomicAdd(a + 2, nrm * g.z);
        }
        __syncthreads();   // protect buffer 'par' before it is refilled at t+2
    }
}

// cur[i].k = relu(Wt[k]*acc[i].k + Vt[k]*x[i] + bt[k]); clears acc for next layer
__global__ void arma_update_kernel(const float* __restrict__ x,
                                   float4* __restrict__ acc,
                                   const float* __restrict__ Wt,
                                   const float* __restrict__ Vt,
                                   const float* __restrict__ bt,
                                   float4* __restrict__ cur, int n) {
    for (long long i = GSTART; i < n; i += GSTRIDE) {
        float4 a = acc[i];
        acc[i] = make_float4(0.0f, 0.0f, 0.0f, 0.0f);
        float xi = x[i];
        float4 o;
        o.x = fmaxf(fmaf(Wt[0], a.x, fmaf(Vt[0], xi, bt[0])), 0.0f);
        o.y = fmaxf(fmaf(Wt[1], a.y, fmaf(Vt[1], xi, bt[1])), 0.0f);
        o.z = fmaxf(fmaf(Wt[2], a.z, fmaf(Vt[2], xi, bt[2])), 0.0f);
        o.w = 0.0f;
        cur[i] = o;
    }
}

// out[i] = sigmoid( mean_k(cur[i].k) * lin_w + lin_b )
__global__ void arma_final_kernel(const float4* __restrict__ cur,
                                  const float* __restrict__ lw,
                                  const float* __restrict__ lb,
                                  float* __restrict__ out, int n) {
    for (long long i = GSTART; i < n; i += GSTRIDE) {
        float4 c = cur[i];
        float m = (c.x + c.y + c.z) * (1.0f / 3.0f);
        float y = fmaf(m, lw[0], lb[0]);
        out[i] = 1.0f / (1.0f + __expf(-y));
    }
}

extern "C" void kernel_launch(void* const* d_in, const int* in_sizes, int n_in,
                              void* d_out, int out_size, void* d_ws, size_t ws_size,
                              hipStream_t stream) {
    const float*     x   = (const float*)d_in[0];       // [N,1] f32
    const long long* ei  = (const long long*)d_in[1];   // [2,E] i64
    const float*     ew  = (const float*)d_in[2];       // [E]   f32
    const float*     w0  = (const float*)d_in[3];       // init_weight [K]
    const float*     wl  = (const float*)d_in[4];       // weight [(T-1)*K]
    const float*     vw  = (const float*)d_in[5];       // root_weight [T*K]
    const float*     bs  = (const float*)d_in[6];       // bias [T*K]
    const float*     lw  = (const float*)d_in[7];       // lin_w [1]
    const float*     lb  = (const float*)d_in[8];       // lin_b [1]
    float*           out = (float*)d_out;

    const int       n = in_sizes[0];                    // N = 1e6
    const long long E = (long long)in_sizes[1] / 2;     // 32e6

    const long long* row = ei;       // edge_index[0]
    const long long* col = ei + E;   // edge_index[1]

    // Workspace (floats): deg/dinv[n] | px[n] | acc[4n] | cur[4n]  = 40 MB
    float* deg = (float*)d_ws;
    float* px  = deg + n;
    float* acc = px + n;
    float* cur = acc + 4ll * n;

    const int nodeBlocks = (n + THREADS - 1) / THREADS;

    // Workspace is poisoned: zero deg, px, acc (contiguous 6n floats).
    arma_zero_kernel<<<2048, THREADS, 0, stream>>>(deg, 6ll * (long long)n);

    // Degrees -> dinv
    arma_deg_kernel<<<EDGE_BLOCKS, THREADS, 0, stream>>>(col, ew, deg, E);
    arma_dinv_kernel<<<nodeBlocks, THREADS, 0, stream>>>(deg, n);

    // Layer 0: P(x) shared across stacks, then per-stack scalar affine + ReLU
    arma_px_kernel<<<EDGE_BLOCKS, THREADS, 0, stream>>>(row, col, ew, deg, x, px, E);
    arma_layer0_kernel<<<nodeBlocks, THREADS, 0, stream>>>(
        x, px, w0, vw + 0, bs + 0, (float4*)cur, n);

    // Layers 1..3: TDM-staged fused-K SpMV + per-node update (re-zeros acc)
    for (int t = 1; t < 4; ++t) {
        arma_spmv3_kernel<<<EDGE_BLOCKS, THREADS, 0, stream>>>(
            row, col, ew, deg, (const float4*)cur, acc, E);
        arma_update_kernel<<<nodeBlocks, THREADS, 0, stream>>>(
            x, (float4*)acc, wl + (t - 1) * 3, vw + t * 3, bs + t * 3, (float4*)cur, n);
    }

    // Mean over stacks -> affine -> sigmoid
    arma_final_kernel<<<nodeBlocks, THREADS, 0, stream>>>((const float4*)cur, lw, lb, out, n);
}